// PerceptionUpdateModule_88845693485749
// MI455X (gfx1250) — compile-verified
//
#include <hip/hip_runtime.h>

#define N_NODES 50000
#define N_EDGES 1600000
#define TSTRIDE 17
#define WPB 4            // waves per block
#define BLOCK_THREADS 128

typedef float v2f __attribute__((ext_vector_type(2)));
typedef float v8f __attribute__((ext_vector_type(8)));

struct LayerDesc {
  const int* ia; const int* ib; const int* op;
  int cin, g, Kpad, Npad, woff;
};
struct NetParams { LayerDesc L[9]; };   // 0-2 fusion, 3-5 attn, 6-8 upd

// ---------------- weight-matrix construction ----------------
__global__ void build_w_kernel(NetParams P, float* __restrict__ W) {
  const float CA[9] = {1.f, 1.f, -1.f, 1.f, 0.f, -1.f, 0.f, 0.f, 0.f};
  const float CB[9] = {1.f, -1.f, 1.f, 0.f, 1.f, 0.f, -1.f, 0.f, 0.f};
  for (int l = 0; l < 9; ++l) {
    LayerDesc d = P.L[l];
    for (int j = (int)threadIdx.x; j < d.Npad; j += (int)blockDim.x) {
      float* col = W + d.woff + j * d.Kpad;      // column-K-major
      for (int k = 0; k < d.Kpad; ++k) col[k] = 0.f;
      if (j < d.g) {
        int op = d.op[j];
        if (op < 7) {
          col[d.ia[j]] += CA[op];
          col[d.ib[j]] += CB[op];
        }
      }
    }
  }
}

__global__ void zero_kernel(float* __restrict__ p, int n) {
  int i = blockIdx.x * blockDim.x + threadIdx.x;
  int stride = gridDim.x * blockDim.x;
  for (; i < n; i += stride) p[i] = 0.f;
}

// ---------------- one gate layer via WMMA f32 16x16x4 ----------------
// src/dst: LDS tiles laid out as [channel*TSTRIDE + row], 16 rows (edges/nodes).
__device__ __forceinline__ void run_layer_wmma(const LayerDesc& d,
                                               const float* __restrict__ W,
                                               const float* src, float* dst,
                                               int lane) {
  const int Mrow = lane & 15;
  const int kh   = (lane >> 4) << 1;   // K offset within fragment: 0 or 2
  const int Moff = (lane >> 4) << 3;   // D row offset: 0 or 8
  const float* Wl = W + d.woff;
  for (int n0 = 0; n0 < d.Npad; n0 += 16) {
    const int col = n0 + Mrow;                 // this lane's output column
    const float* wcol = Wl + col * d.Kpad;
    v8f acc = {0.f, 0.f, 0.f, 0.f, 0.f, 0.f, 0.f, 0.f};
    for (int k0 = 0; k0 < d.Kpad; k0 += 4) {
      v2f a, b;
      a.x = src[(k0 + kh + 0) * TSTRIDE + Mrow];
      a.y = src[(k0 + kh + 1) * TSTRIDE + Mrow];
      b.x = wcol[k0 + kh + 0];
      b.y = wcol[k0 + kh + 1];
      acc = __builtin_amdgcn_wmma_f32_16x16x4_f32(
          false, a, false, b, (short)0, acc, false, false);
    }
    // fixup (max/min gates) + writeback; lane owns one column, 8 rows
    int opc = -1, iac = 0, ibc = 0;
    if (col < d.g) {
      opc = d.op[col];
      if (opc >= 7) { iac = d.ia[col]; ibc = d.ib[col]; }
    }
#pragma unroll
    for (int r = 0; r < 8; ++r) {
      int M = r + Moff;
      float v;
      if (opc < 0) {
        v = 0.f;                                  // padding column
      } else if (opc >= 7) {
        float av = src[iac * TSTRIDE + M];
        float bv = src[ibc * TSTRIDE + M];
        v = (opc == 7) ? fmaxf(av, bv) : fminf(av, bv);
      } else {
        v = acc[r];
      }
      dst[col * TSTRIDE + M] = v;
    }
  }
  __syncthreads();
}

// ---------------- edge pass: attn -> w, fusion -> fused, atomic accumulate ----------------
__global__ void __launch_bounds__(BLOCK_THREADS)
edge_kernel(const float* __restrict__ ns, const float* __restrict__ einfo,
            const int* __restrict__ esrc, const int* __restrict__ edst,
            NetParams P, const float* __restrict__ W,
            float* __restrict__ abs_sum, float* __restrict__ agg) {
  __shared__ float t0[WPB][64 * TSTRIDE];
  __shared__ float t1[WPB][64 * TSTRIDE];
  __shared__ float wL[WPB][16];
  __shared__ float duL[WPB][16];
  __shared__ float dvL[WPB][16];
  __shared__ int   srcL[WPB][16];

  const int lane = threadIdx.x & 31;
  const int wv   = threadIdx.x >> 5;
  const int e0   = (blockIdx.x * WPB + wv) * 16;   // N_EDGES % 64 == 0, no tail

  if (lane < 16) {
    int e = e0 + lane;
    int s = esrc[e], dd = edst[e];
    float du = ns[s], dv = ns[dd];
    t0[wv][0 * TSTRIDE + lane] = du;
    t0[wv][1 * TSTRIDE + lane] = dv;
    t0[wv][2 * TSTRIDE + lane] = einfo[2 * e + 0];
    t0[wv][3 * TSTRIDE + lane] = einfo[2 * e + 1];
    srcL[wv][lane] = s; duL[wv][lane] = du; dvL[wv][lane] = dv;
  }
  __syncthreads();

  // attn net: 4 -> 64 -> 32 -> 1
  run_layer_wmma(P.L[3], W, t0[wv], t1[wv], lane);
  run_layer_wmma(P.L[4], W, t1[wv], t0[wv], lane);
  run_layer_wmma(P.L[5], W, t0[wv], t1[wv], lane);

  if (lane < 16) wL[wv][lane] = t1[wv][0 * TSTRIDE + lane];
  __syncthreads();
  if (lane < 16) {
    t0[wv][0 * TSTRIDE + lane] = duL[wv][lane];
    t0[wv][1 * TSTRIDE + lane] = dvL[wv][lane];
    t0[wv][2 * TSTRIDE + lane] = 0.f;
    t0[wv][3 * TSTRIDE + lane] = 0.f;
  }
  __syncthreads();

  // fusion net: 2 -> 64 -> 64 -> 32
  run_layer_wmma(P.L[0], W, t0[wv], t1[wv], lane);
  run_layer_wmma(P.L[1], W, t1[wv], t0[wv], lane);
  run_layer_wmma(P.L[2], W, t0[wv], t1[wv], lane);

  // accumulate: abs_sum[src] += |w|;  agg[src][c] += fused*w  (normalization deferred)
  if (lane < 16) atomicAdd(&abs_sum[srcL[wv][lane]], fabsf(wL[wv][lane]));
#pragma unroll
  for (int i = 0; i < 16; ++i) {
    int M = i;          // edge row within tile
    int c = lane;       // channel 0..31
    float v = t1[wv][c * TSTRIDE + M] * wL[wv][M];
    atomicAdd(&agg[(long)srcL[wv][M] * 32 + c], v);
  }
}

// ---------------- node pass: normalize agg, upd net, residual ----------------
__global__ void __launch_bounds__(BLOCK_THREADS)
node_kernel(const float* __restrict__ ns, NetParams P, const float* __restrict__ W,
            const float* __restrict__ abs_sum, const float* __restrict__ agg,
            float* __restrict__ out) {
  __shared__ float t0[WPB][64 * TSTRIDE];
  __shared__ float t1[WPB][64 * TSTRIDE];
  const int lane = threadIdx.x & 31;
  const int wv   = threadIdx.x >> 5;
  const int n0   = (blockIdx.x * WPB + wv) * 16;
  const bool active = (n0 < N_NODES);   // N_NODES % 16 == 0 -> whole-wave valid
  float s = 0.f;
  if (lane < 16) {
    int n = active ? (n0 + lane) : 0;
    s = ns[n];
    float denom = fmaxf(abs_sum[n], 1e-6f);
    t0[wv][0 * TSTRIDE + lane] = s;
#pragma unroll
    for (int c = 0; c < 32; ++c)
      t0[wv][(c + 1) * TSTRIDE + lane] = agg[(long)n * 32 + c] / denom;
    t0[wv][33 * TSTRIDE + lane] = 0.f;   // K padding to 36
    t0[wv][34 * TSTRIDE + lane] = 0.f;
    t0[wv][35 * TSTRIDE + lane] = 0.f;
  }
  __syncthreads();

  // upd net: 33 -> 64 -> 64 -> 1
  run_layer_wmma(P.L[6], W, t0[wv], t1[wv], lane);
  run_layer_wmma(P.L[7], W, t1[wv], t0[wv], lane);
  run_layer_wmma(P.L[8], W, t0[wv], t1[wv], lane);

  if (active && lane < 16)
    out[n0 + lane] = s + t1[wv][0 * TSTRIDE + lane];
}

// ---------------- host side ----------------
static inline int rup(int x, int m) { return ((x + m - 1) / m) * m; }

extern "C" void kernel_launch(void* const* d_in, const int* in_sizes, int n_in,
                              void* d_out, int out_size, void* d_ws, size_t ws_size,
                              hipStream_t stream) {
  (void)in_sizes; (void)n_in; (void)out_size; (void)ws_size;
  const float* ns    = (const float*)d_in[0];
  const float* einfo = (const float*)d_in[1];
  const int*   esrc  = (const int*)d_in[2];
  const int*   edst  = (const int*)d_in[3];

  // gate-net dims in d_in order: fusion(ia,ib,op)x3 @4..12, attn @13..21, upd @22..30
  const int gs[9]  = {64, 64, 32, 64, 32, 1, 64, 64, 1};
  const int cins[9]= { 2, 64, 64,  4, 64, 32, 33, 64, 64};

  NetParams P;
  int woff = 0;
  for (int l = 0; l < 9; ++l) {
    int base = 4 + l * 3;
    P.L[l].ia  = (const int*)d_in[base + 0];
    P.L[l].ib  = (const int*)d_in[base + 1];
    P.L[l].op  = (const int*)d_in[base + 2];
    P.L[l].cin = cins[l];
    P.L[l].g   = gs[l];
    P.L[l].Kpad = rup(cins[l], 4);
    P.L[l].Npad = rup(gs[l], 16);
    P.L[l].woff = woff;
    woff += P.L[l].Kpad * P.L[l].Npad;
  }

  float* wsf     = (float*)d_ws;
  float* abs_sum = wsf;                       // N_NODES
  float* agg     = wsf + N_NODES;             // N_NODES * 32
  float* W       = wsf + N_NODES + (long)N_NODES * 32;   // 16640 floats

  const int nzero = N_NODES + N_NODES * 32;
  zero_kernel<<<2048, 256, 0, stream>>>(wsf, nzero);
  build_w_kernel<<<1, 256, 0, stream>>>(P, W);

  const int edge_blocks = N_EDGES / (WPB * 16);                 // 25000 exact
  edge_kernel<<<edge_blocks, BLOCK_THREADS, 0, stream>>>(
      ns, einfo, esrc, edst, P, W, abs_sum, agg);

  const int node_waves  = (N_NODES + 15) / 16;                  // 3125
  const int node_blocks = (node_waves + WPB - 1) / WPB;         // 782
  node_kernel<<<node_blocks, BLOCK_THREADS, 0, stream>>>(
      ns, P, W, abs_sum, agg, (float*)d_out);
}